// VowelHarmonyConstraint_79663053406447
// MI455X (gfx1250) — compile-verified
//
#include <hip/hip_runtime.h>
#include <cstdint>

// VowelHarmonyConstraint — MI455X (gfx1250) streaming kernel.
// Memory-bound (~300 MB @ 23.3 TB/s => ~13us floor). One wave32 per row:
// 32 lanes x 8 floats = V=256. CDNA5 paths used: async global->LDS staging of
// next row's prev-chars (ASYNCcnt double buffer), s_wait_asynccnt, NT cache
// policy on the 268 MB single-use stream, global_prefetch, LDS gathers,
// wave32 butterfly reductions. No WMMA: 0.5 FLOP/byte, matrix ops useless.

typedef float v4f __attribute__((ext_vector_type(4)));

#define VOCAB 256
#define SEQ   64
#define WPB   8              // waves per block
#define BLOCK (WPB * 32)

__global__ __launch_bounds__(BLOCK) void vowel_harmony_kernel(
    const float* __restrict__ dist,   // [B, 256]
    const int*   __restrict__ prev,   // [B, 64]
    const float* __restrict__ fmask,  // [256]
    const float* __restrict__ bmask,  // [256]
    float*       __restrict__ out,    // [B, 256]
    int nrows)
{
    __shared__ float s_fm[VOCAB];
    __shared__ float s_bm[VOCAB];
    __shared__ int   s_prev[WPB][2][SEQ];   // per-wave double buffer (async dest)

    const int tid  = threadIdx.x;
    const int lane = tid & 31;
    const int wave = tid >> 5;

    // Char-indexed mask copies in LDS for the count-phase gathers.
    if (tid < VOCAB) { s_fm[tid] = fmask[tid]; s_bm[tid] = bmask[tid]; }

    // Each lane permanently owns vocab slots [lane*8, lane*8+8): cache its
    // mask values in registers so pass-2 weights need no LDS traffic.
    float fm8[8], bm8[8];
#pragma unroll
    for (int j = 0; j < 8; ++j) {
        fm8[j] = fmask[lane * 8 + j];
        bm8[j] = bmask[lane * 8 + j];
    }
    __syncthreads();

    const int gw   = blockIdx.x * WPB + wave;    // global wave id
    const int step = gridDim.x * WPB;

    // LDS byte offsets of this wave's two staging buffers (generic LDS
    // pointers carry the DS-space offset in their low 32 bits).
    const uint32_t lds0 = (uint32_t)(uintptr_t)(&s_prev[wave][0][0]);
    const uint32_t lds1 = (uint32_t)(uintptr_t)(&s_prev[wave][1][0]);

    int row = gw;
    if (row >= nrows) return;

    // Kick off async staging of the first row's prev-chars (2 ints/lane = b64:
    // 32 lanes x 8B = the whole 256B row in one async op).
    {
        const int* g = prev + (size_t)row * SEQ + lane * 2;
        asm volatile("global_load_async_to_lds_b64 %0, %1, off"
                     :: "v"(lds0 + (uint32_t)(lane * 8)),
                        "v"((uint64_t)(uintptr_t)g)
                     : "memory");
    }

    int buf = 0;
    for (; row < nrows; row += step, buf ^= 1) {
        // ---- prefetch / async-stage one row ahead -------------------------
        const int nrow = row + step;
        const int prow = (nrow < nrows) ? nrow : row;   // clamped dummy reload
        __builtin_prefetch(dist + (size_t)prow * VOCAB + lane * 8, 0, 0);
        {
            const int* g = prev + (size_t)prow * SEQ + lane * 2;
            const uint32_t l = (buf ? lds0 : lds1) + (uint32_t)(lane * 8);
            asm volatile("global_load_async_to_lds_b64 %0, %1, off"
                         :: "v"(l), "v"((uint64_t)(uintptr_t)g)
                         : "memory");
        }

        // ---- stream this row's distribution (read-once => non-temporal) ---
        const float* drow = dist + (size_t)row * VOCAB;
        v4f d0 = __builtin_nontemporal_load((const v4f*)(drow + lane * 8));
        v4f d1 = __builtin_nontemporal_load((const v4f*)(drow + lane * 8 + 4));

        // Wait for the async load issued LAST iteration; completions are
        // in-order, so asynccnt<=1 leaves only the just-issued one pending.
        asm volatile("s_wait_asynccnt 0x1" ::: "memory");

        // ---- harmony counts: gather masks by char, butterfly-reduce -------
        const int c0 = s_prev[wave][buf][lane]      & 255;
        const int c1 = s_prev[wave][buf][lane + 32] & 255;
        float cf = s_fm[c0] + s_fm[c1];
        float cb = s_bm[c0] + s_bm[c1];
#pragma unroll
        for (int m = 16; m >= 1; m >>= 1) {
            cf += __shfl_xor(cf, m, 32);
            cb += __shfl_xor(cb, m, 32);
        }
        // softmax([cf, cb])
        const float pf = 1.0f / (1.0f + expf(cb - cf));
        const float pb = 1.0f - pf;

        // ---- reweight + row sum -------------------------------------------
        v4f o0, o1;
        float psum = 0.0f;
#pragma unroll
        for (int j = 0; j < 4; ++j) {
            const float w = 0.5f + 0.5f * (pf * fm8[j] + pb * bm8[j]);
            o0[j] = d0[j] * w;
            psum += o0[j];
        }
#pragma unroll
        for (int j = 0; j < 4; ++j) {
            const float w = 0.5f + 0.5f * (pf * fm8[4 + j] + pb * bm8[4 + j]);
            o1[j] = d1[j] * w;
            psum += o1[j];
        }
#pragma unroll
        for (int m = 16; m >= 1; m >>= 1) psum += __shfl_xor(psum, m, 32);
        const float inv = 1.0f / (psum + 1e-10f);

        // ---- normalize + NT store (never re-read) -------------------------
        float* orow = out + (size_t)row * VOCAB;
        __builtin_nontemporal_store(o0 * inv, (v4f*)(orow + lane * 8));
        __builtin_nontemporal_store(o1 * inv, (v4f*)(orow + lane * 8 + 4));
    }
}

extern "C" void kernel_launch(void* const* d_in, const int* in_sizes, int n_in,
                              void* d_out, int out_size, void* d_ws, size_t ws_size,
                              hipStream_t stream) {
    const float* dist = (const float*)d_in[0];   // output_distribution [B,256] f32
    const int*   prev = (const int*)d_in[1];     // previous_chars     [B,64]  i32
    const float* fm   = (const float*)d_in[2];   // front_mask [256] f32
    const float* bm   = (const float*)d_in[3];   // back_mask  [256] f32
    float* out = (float*)d_out;

    const int nrows = in_sizes[0] / VOCAB;       // B = 131072
    const int grid  = 4096;                      // 32768 waves, 4 rows/wave
    vowel_harmony_kernel<<<grid, BLOCK, 0, stream>>>(dist, prev, fm, bm, out, nrows);
}